// CrossCoderDecoder_74534862455448
// MI455X (gfx1250) — compile-verified
//
#include <hip/hip_runtime.h>

typedef float v2f __attribute__((ext_vector_type(2)));
typedef float v8f __attribute__((ext_vector_type(8)));

#define B_      32
#define L_      2
#define F_      65536
#define D_      768
#define N_      (L_ * D_)        // 1536 output columns
#define SPLITK  64
#define KR      (F_ / SPLITK)    // 1024 K per block
#define KC      256              // K per LDS stage
#define LDSROW  (KC + 4)         // +4 floats pad: bank stride 4, float4-aligned rows
#define TPB     128              // 4 waves; block tile = 32 x 128

// Partial GEMM: each block produces a 32x128 fp32 partial tile for one K-split.
// Each wave owns a 32x32 tile: 4 f32 WMMA accumulators, A fragments shared
// across the two 16-column groups.
__global__ __launch_bounds__(TPB)
void crosscoder_wmma_partial(const float* __restrict__ f,
                             const float* __restrict__ weight,
                             float* __restrict__ partial) {
    __shared__ float fs[B_ * LDSROW];

    const int nblk = blockIdx.x;       // 0..11 : 128-column tile
    const int s    = blockIdx.y;       // 0..63 : K split
    const int tid  = threadIdx.x;
    const int wave = tid >> 5;
    const int lane = tid & 31;
    const int half = lane >> 4;        // 0 | 1
    const int m    = lane & 15;        // row-within-16 / col-within-16
    const int kr0  = half * 2;         // K row offset held by .x (ISA A/B layout)

    const int n0    = nblk * 128 + wave * 32;   // first column of this wave's tile
    const int col0  = n0 + m;                   // lane's column in group 0 (+16 for group 1)
    const int l     = n0 / D_;                  // tile never crosses l (768 = 6*128)
    const int d     = col0 - l * D_;
    const int kbase = s * KR;

    const float* wp = weight + (size_t)l * F_ * D_ + (size_t)kbase * D_ + d;
    const float* fp = f + kbase;

    v8f acc00 = {};   // batches 0..15,  cols n0..n0+15
    v8f acc01 = {};   // batches 0..15,  cols n0+16..n0+31
    v8f acc10 = {};   // batches 16..31, cols n0..n0+15
    v8f acc11 = {};   // batches 16..31, cols n0+16..n0+31

    for (int kc = 0; kc < KR; kc += KC) {
        __syncthreads();
        // Stage f[0..31][kbase+kc .. +KC) into LDS, coalesced float4 copies.
        #pragma unroll
        for (int it = 0; it < (B_ * KC / 4) / TPB; ++it) {
            int i  = tid + it * TPB;
            int b  = i >> 6;            // KC/4 = 64 float4 per row
            int kq = i & 63;
            float4 v = *(const float4*)(fp + (size_t)b * F_ + kc + kq * 4);
            *(float4*)&fs[b * LDSROW + kq * 4] = v;
        }
        __syncthreads();

        // Speculative prefetch of next weight chunk, hoisted so the inner
        // loop stays a single unrolled body (no loop splitting).
        if (kc + KC < KR) {
            #pragma unroll
            for (int j = 0; j < 4; ++j)
                __builtin_prefetch(wp + (size_t)(kc + KC + j * 64) * D_, 0, 1);
        }

        #pragma unroll 8
        for (int k = 0; k < KC; k += 4) {
            // A fragments from LDS (16x4 fp32, lane m holds K=kr0..kr0+1)
            v2f a0 = *(const v2f*)&fs[m * LDSROW + k + kr0];
            v2f a1 = *(const v2f*)&fs[(m + 16) * LDSROW + k + kr0];
            // B fragments from global: row K=kr0+v striped over 16 lanes.
            // Column group 1 is the same address +64B -> shared base register.
            const float* wk = wp + (size_t)(kc + k + kr0) * D_;
            v2f b0, b1;
            b0.x = wk[0];        b0.y = wk[D_];
            b1.x = wk[16];       b1.y = wk[D_ + 16];

            acc00 = __builtin_amdgcn_wmma_f32_16x16x4_f32(
                false, a0, false, b0, (short)0, acc00, false, false);
            acc10 = __builtin_amdgcn_wmma_f32_16x16x4_f32(
                false, a1, false, b0, (short)0, acc10, false, false);
            acc01 = __builtin_amdgcn_wmma_f32_16x16x4_f32(
                false, a0, false, b1, (short)0, acc01, false, false);
            acc11 = __builtin_amdgcn_wmma_f32_16x16x4_f32(
                false, a1, false, b1, (short)0, acc11, false, false);
        }
    }

    // C layout: VGPR r, lanes 0-15 -> M=r ; lanes 16-31 -> M=r+8 ; N = lane&15
    float* pout = partial + (size_t)s * (B_ * N_);
    #pragma unroll
    for (int r = 0; r < 8; ++r) {
        int row = r + 8 * half;
        pout[(size_t)row        * N_ + col0]      = acc00[r];
        pout[(size_t)row        * N_ + col0 + 16] = acc01[r];
        pout[(size_t)(row + 16) * N_ + col0]      = acc10[r];
        pout[(size_t)(row + 16) * N_ + col0 + 16] = acc11[r];
    }
}

// Deterministic reduction: out = bias + sum over SPLITK partials.
__global__ __launch_bounds__(256)
void crosscoder_reduce(const float* __restrict__ partial,
                       const float* __restrict__ bias,
                       float* __restrict__ out) {
    int i = blockIdx.x * 256 + threadIdx.x;  // 0 .. B_*N_-1
    int r = i % N_;                          // l*768 + d  (bias is (L,D) flat)
    float acc = bias[r];
    #pragma unroll 8
    for (int s = 0; s < SPLITK; ++s)
        acc += partial[(size_t)s * (B_ * N_) + i];
    out[i] = acc;
}

extern "C" void kernel_launch(void* const* d_in, const int* in_sizes, int n_in,
                              void* d_out, int out_size, void* d_ws, size_t ws_size,
                              hipStream_t stream) {
    const float* f      = (const float*)d_in[0];   // (32, 65536)
    const float* weight = (const float*)d_in[1];   // (2, 65536, 768)
    const float* bias   = (const float*)d_in[2];   // (2, 768)
    float*       out    = (float*)d_out;           // (32, 2, 768)
    float*       part   = (float*)d_ws;            // SPLITK * 32 * 1536 fp32 ~ 12.6 MB

    dim3 grid(N_ / 128, SPLITK);
    crosscoder_wmma_partial<<<grid, TPB, 0, stream>>>(f, weight, part);
    crosscoder_reduce<<<(B_ * N_ + 255) / 256, 256, 0, stream>>>(part, bias, out);
}